// model_2482491097864
// MI455X (gfx1250) — compile-verified
//
#include <hip/hip_runtime.h>
#include <hip/hip_bf16.h>

// ---------------- problem constants (match reference) ----------------
constexpr int N_    = 50000;   // nodes
constexpr int N_PAD = 50048;   // nodes padded to multiple of 64 (ws buffers only)
constexpr int E_    = 800000;  // edges (divisible by 64)
constexpr int FIN   = 8;       // node feature dim
constexpr int S_    = 6;       // edge feature dim
constexpr int H_    = 20;      // ECC hidden
constexpr int NOUT  = 20;      // ECC out channels
constexpr int G_    = 32;      // GCN channels
constexpr int D1_   = 16;      // dense1
constexpr int NG    = 512;     // graphs

typedef __attribute__((ext_vector_type(2))) float v2f;
typedef __attribute__((ext_vector_type(8))) float v8f;

// fp32 WMMA: D = A(16x4) * B(4x16) + C(16x16)
__device__ __forceinline__ v8f wmma4(v2f a, v2f b, v8f c) {
    return __builtin_amdgcn_wmma_f32_16x16x4_f32(
        false, a, false, b, (short)0, c, false, false);
}

// =====================================================================
// Kernel 1: fused edge pipeline (fully branch-free inner stages).
// =====================================================================
#define EW 4   // waves per block -> 64 edges per block, E_ % 64 == 0
__global__ __launch_bounds__(EW * 32) void edge_kernel(
    const float* __restrict__ e, const int* __restrict__ eidx,
    const float* __restrict__ x,
    const float* __restrict__ W1, const float* __restrict__ b1,
    const float* __restrict__ W2, const float* __restrict__ b2,
    const float* __restrict__ W3, const float* __restrict__ b3,
    float* __restrict__ agg, float* __restrict__ deg)
{
    __shared__ float hbuf[EW][16][32];   // hidden tile, cols 20..31 are exact zeros
    __shared__ float wbuf[EW][16][161];  // per-edge filter tile (160 + pad)
    __shared__ float w1p[8][32];         // W1 zero-padded [6x20]->[8x32]
    __shared__ float w2p[20][32];        // W2 zero-padded [20x20]->[20x32]
    __shared__ float b1p[32], b2p[32];

    const int tid  = threadIdx.x;
    const int lane = tid & 31;
    const int wave = tid >> 5;
    const int half = lane >> 4;
    const int lm   = lane & 15;
    const int eBase = (blockIdx.x * EW + wave) * 16;   // always full tile

    // ---- cooperative, branch-free weight staging (clamped addr + select) ----
    for (int i = tid; i < 8 * 32; i += EW * 32) {
        const int k = i >> 5, c = i & 31;
        const float v = W1[(k < S_ ? k : S_ - 1) * H_ + (c < H_ ? c : H_ - 1)];
        w1p[k][c] = (k < S_ && c < H_) ? v : 0.f;
    }
    for (int i = tid; i < 20 * 32; i += EW * 32) {
        const int k = i >> 5, c = i & 31;
        const float v = W2[k * H_ + (c < H_ ? c : H_ - 1)];
        w2p[k][c] = (c < H_) ? v : 0.f;
    }
    if (tid < 32) {
        const float v1 = b1[tid < H_ ? tid : H_ - 1];
        const float v2 = b2[tid < H_ ? tid : H_ - 1];
        b1p[tid] = (tid < H_) ? v1 : 0.f;
        b2p[tid] = (tid < H_) ? v2 : 0.f;
    }

    // prefetch next tile of edge features
    if ((eBase + EW * 16) < E_)
        __builtin_prefetch(e + (size_t)(eBase + EW * 16) * S_ + lane, 0, 1);

    __syncthreads();

    // ---------------- stage 1: h1 = relu(e @ W1 + b1), K=6 pad 8 ----------
    v8f acc0, acc1;
    {
        const int er = eBase + lm;               // always < E_
        v8f accs[2];
        #pragma unroll
        for (int nt = 0; nt < 2; ++nt) {
            v8f c = {};
            #pragma unroll
            for (int k0 = 0; k0 < 8; k0 += 4) {
                const int k = k0 + 2 * half;     // <= 7
                v2f a, b;
                const float ax = e[(size_t)er * S_ + (k     < S_ ? k     : S_ - 1)];
                const float ay = e[(size_t)er * S_ + (k + 1 < S_ ? k + 1 : S_ - 1)];
                a.x = (k     < S_) ? ax : 0.f;
                a.y = (k + 1 < S_) ? ay : 0.f;
                b.x = w1p[k][nt * 16 + lm];
                b.y = w1p[k + 1][nt * 16 + lm];
                c = wmma4(a, b, c);
            }
            accs[nt] = c;
        }
        acc0 = accs[0]; acc1 = accs[1];
    }
    // unconditional stores: cols 20..31 are relu(0+0)=0 by zero-padded B/bias
    #pragma unroll
    for (int r = 0; r < 8; ++r) {
        const int row = r + 8 * half;
        hbuf[wave][row][lm]      = fmaxf(acc0[r] + b1p[lm],      0.f);
        hbuf[wave][row][16 + lm] = fmaxf(acc1[r] + b1p[16 + lm], 0.f);
    }
    __syncthreads();

    // ---------------- stage 2: h2 = relu(h1 @ W2 + b2), K=20 --------------
    {
        v8f accs[2];
        #pragma unroll
        for (int nt = 0; nt < 2; ++nt) {
            v8f c = {};
            #pragma unroll
            for (int k0 = 0; k0 < H_; k0 += 4) {
                const int k = k0 + 2 * half;     // <= 19
                v2f a, b;
                a.x = hbuf[wave][lm][k];
                a.y = hbuf[wave][lm][k + 1];
                b.x = w2p[k][nt * 16 + lm];
                b.y = w2p[k + 1][nt * 16 + lm];
                c = wmma4(a, b, c);
            }
            accs[nt] = c;
        }
        acc0 = accs[0]; acc1 = accs[1];
    }
    __syncthreads();
    #pragma unroll
    for (int r = 0; r < 8; ++r) {
        const int row = r + 8 * half;
        hbuf[wave][row][lm]      = fmaxf(acc0[r] + b2p[lm],      0.f);
        hbuf[wave][row][16 + lm] = fmaxf(acc1[r] + b2p[16 + lm], 0.f);
    }
    __syncthreads();

    // ------------- stage 3: w_edge = h2 @ W3 + b3, [16x20]@[20x160] -------
    #pragma unroll
    for (int nt = 0; nt < 10; ++nt) {
        v8f c = {};
        const int cN = nt * 16 + lm;             // < 160 always
        #pragma unroll
        for (int k0 = 0; k0 < H_; k0 += 4) {
            const int k = k0 + 2 * half;
            v2f a, b;
            a.x = hbuf[wave][lm][k];
            a.y = hbuf[wave][lm][k + 1];
            b.x = W3[k       * 160 + cN];
            b.y = W3[(k + 1) * 160 + cN];
            c = wmma4(a, b, c);
        }
        const float bb = b3[cN];
        #pragma unroll
        for (int r = 0; r < 8; ++r)
            wbuf[wave][r + 8 * half][cN] = c[r] + bb;
    }
    __syncthreads();

    // ---------------- stage 4: einsum + scatter (no edge guards) ----------
    #pragma unroll
    for (int j = 0; j < 10; ++j) {
        const int idx = lane * 10 + j;           // 0..319
        const int el  = idx / NOUT;
        const int o   = idx % NOUT;
        const int eg  = eBase + el;
        const int rn  = eidx[eg];
        const int cn  = eidx[E_ + eg];
        float m = 0.f;
        #pragma unroll
        for (int f = 0; f < FIN; ++f)
            m += x[cn * FIN + f] * wbuf[wave][el][f * NOUT + o];
        atomicAdd(&agg[rn * NOUT + o], m);
    }
    if (lane < 16)
        atomicAdd(&deg[eidx[eBase + lane]], 1.0f);
}

// =====================================================================
// Kernel 2: x1 = relu(agg + x @ ecc_root + ecc_b)   [16x8]@[8x20] WMMA
//   x1 stored with stride 32 (cols 20..31 garbage, never read).
// =====================================================================
__global__ __launch_bounds__(128) void x1_kernel(
    const float* __restrict__ x, const float* __restrict__ Wr,
    const float* __restrict__ br, const float* __restrict__ agg,
    float* __restrict__ x1)
{
    __shared__ float wrp[8][32];   // ecc_root zero-padded [8x20]->[8x32]
    __shared__ float brp[32];

    const int tid  = threadIdx.x;
    const int lane = tid & 31;
    const int wave = tid >> 5;
    const int half = lane >> 4;
    const int lm   = lane & 15;
    const int nBase = (blockIdx.x * 4 + wave) * 16;

    for (int i = tid; i < 8 * 32; i += 128) {
        const int k = i >> 5, c = i & 31;
        const float v = Wr[k * NOUT + (c < NOUT ? c : NOUT - 1)];
        wrp[k][c] = (c < NOUT) ? v : 0.f;
    }
    if (tid < 32) {
        const float v = br[tid < NOUT ? tid : NOUT - 1];
        brp[tid] = (tid < NOUT) ? v : 0.f;
    }
    __syncthreads();

    const int nr  = nBase + lm;
    const int nrc = (nr < N_) ? nr : (N_ - 1);   // clamped input row

    #pragma unroll
    for (int nt = 0; nt < 2; ++nt) {
        v8f c = {};
        #pragma unroll
        for (int k0 = 0; k0 < FIN; k0 += 4) {
            const int k = k0 + 2 * half;         // <= 7
            v2f a, b;
            a.x = x[(size_t)nrc * FIN + k];
            a.y = x[(size_t)nrc * FIN + k + 1];
            b.x = wrp[k][nt * 16 + lm];
            b.y = wrp[k + 1][nt * 16 + lm];
            c = wmma4(a, b, c);
        }
        const int cN  = nt * 16 + lm;
        const int cNc = (cN < NOUT) ? cN : NOUT - 1;
        #pragma unroll
        for (int r = 0; r < 8; ++r) {
            const int row = nBase + r + 8 * half;   // < N_PAD always
            x1[(size_t)row * 32 + cN] =
                fmaxf(c[r] + agg[(size_t)row * NOUT + cNc] + brp[cN], 0.f);
        }
    }
}

// =====================================================================
// Kernel 3: hw = hin @ W   [N_PAD,Cin]@[Cin,32]  — fully guard-free WMMA
// =====================================================================
template <int CIN, int LDA>
__global__ __launch_bounds__(128) void gemm_node(
    const float* __restrict__ hin, const float* __restrict__ W,
    float* __restrict__ hw)
{
    const int lane = threadIdx.x & 31;
    const int wave = threadIdx.x >> 5;
    const int half = lane >> 4;
    const int lm   = lane & 15;
    const int nBase = (blockIdx.x * 4 + wave) * 16;   // < N_PAD
    const int nr = nBase + lm;

    #pragma unroll
    for (int nt = 0; nt < 2; ++nt) {
        v8f c = {};
        const int cN = nt * 16 + lm;
        #pragma unroll
        for (int k0 = 0; k0 < CIN; k0 += 4) {
            const int k = k0 + 2 * half;          // < CIN (CIN % 4 == 0)
            v2f a, b;
            a.x = hin[(size_t)nr * LDA + k];
            a.y = hin[(size_t)nr * LDA + k + 1];
            b.x = W[k       * G_ + cN];
            b.y = W[(k + 1) * G_ + cN];
            c = wmma4(a, b, c);
        }
        #pragma unroll
        for (int r = 0; r < 8; ++r) {
            const int row = nBase + r + 8 * half; // < N_PAD
            hw[(size_t)row * G_ + cN] = c[r];
        }
    }
}

// ------------------- VALU helper kernels -------------------
__global__ void fill_kernel(float* __restrict__ p, float v, int n) {
    int i = blockIdx.x * blockDim.x + threadIdx.x;
    if (i < n) p[i] = v;
}

__global__ void rsqrt_kernel(const float* __restrict__ deg, float* __restrict__ dinv, int n) {
    int i = blockIdx.x * blockDim.x + threadIdx.x;
    if (i < n) dinv[i] = rsqrtf(deg[i]);
}

__global__ void gcn_scatter(const int* __restrict__ eidx, const float* __restrict__ dinv,
                            const float* __restrict__ hw, float* __restrict__ nmsg)
{
    int eg = blockIdx.x * blockDim.x + threadIdx.x;
    if (eg >= E_) return;
    const int rn = eidx[eg];
    const int cn = eidx[E_ + eg];
    const float en = dinv[rn] * dinv[cn];
    #pragma unroll
    for (int g = 0; g < G_; ++g)
        atomicAdd(&nmsg[(size_t)rn * G_ + g], en * hw[(size_t)cn * G_ + g]);
}

__global__ void gcn_finalize(const float* __restrict__ nmsg, const float* __restrict__ hw,
                             const float* __restrict__ dinv, const float* __restrict__ b,
                             float* __restrict__ out)
{
    int i = blockIdx.x * blockDim.x + threadIdx.x;
    if (i >= N_ * G_) return;
    const int n = i >> 5, g = i & 31;
    const float sn = dinv[n] * dinv[n];
    out[i] = fmaxf(nmsg[i] + sn * hw[i] + b[g], 0.f);
}

__global__ void pool_kernel(const float* __restrict__ x3, const int* __restrict__ seg,
                            float* __restrict__ pooled)
{
    int i = blockIdx.x * blockDim.x + threadIdx.x;
    if (i >= N_ * G_) return;
    const int n = i >> 5, g = i & 31;
    atomicAdd(&pooled[(size_t)seg[n] * G_ + g], x3[i]);
}

__global__ void head_kernel(const float* __restrict__ pooled,
                            const float* __restrict__ d1w, const float* __restrict__ d1b,
                            const float* __restrict__ d2w, const float* __restrict__ d2b,
                            float* __restrict__ out)
{
    int gr = blockIdx.x * blockDim.x + threadIdx.x;
    if (gr >= NG) return;
    float acc = d2b[0];
    #pragma unroll
    for (int d = 0; d < D1_; ++d) {
        float s = d1b[d];
        #pragma unroll
        for (int g = 0; g < G_; ++g)
            s += pooled[gr * G_ + g] * d1w[g * D1_ + d];
        acc += s * d2w[d];
    }
    out[gr] = acc;
}

// =====================================================================
extern "C" void kernel_launch(void* const* d_in, const int* in_sizes, int n_in,
                              void* d_out, int out_size, void* d_ws, size_t ws_size,
                              hipStream_t stream) {
    const float* x    = (const float*)d_in[0];
    const float* e    = (const float*)d_in[1];
    const int*   eidx = (const int*)  d_in[2];
    const int*   seg  = (const int*)  d_in[3];
    const float* W1   = (const float*)d_in[4];
    const float* b1   = (const float*)d_in[5];
    const float* W2   = (const float*)d_in[6];
    const float* b2   = (const float*)d_in[7];
    const float* W3   = (const float*)d_in[8];
    const float* b3   = (const float*)d_in[9];
    const float* Wr   = (const float*)d_in[10];
    const float* br   = (const float*)d_in[11];
    const float* g1w  = (const float*)d_in[12];
    const float* g1b  = (const float*)d_in[13];
    const float* g2w  = (const float*)d_in[14];
    const float* g2b  = (const float*)d_in[15];
    const float* d1w  = (const float*)d_in[16];
    const float* d1b  = (const float*)d_in[17];
    const float* d2w  = (const float*)d_in[18];
    const float* d2b  = (const float*)d_in[19];
    float* out = (float*)d_out;

    // workspace layout (floats), node buffers padded to N_PAD rows
    float* ws     = (float*)d_ws;
    float* agg    = ws;                     // N_PAD*20
    float* deg    = agg  + N_PAD * NOUT;    // N_PAD
    float* dinv   = deg  + N_PAD;           // N_PAD
    float* x1     = dinv + N_PAD;           // N_PAD*32 (stride 32, 20 valid)
    float* hw     = x1   + N_PAD * 32;      // N_PAD*32
    float* nmsg   = hw   + N_PAD * G_;      // N_PAD*32
    float* x2     = nmsg + N_PAD * G_;      // N_PAD*32
    float* x3     = x2   + N_PAD * G_;      // N_PAD*32
    float* pooled = x3   + N_PAD * G_;      // 512*32

    const int TB = 256;
    auto blocks = [](int n, int tb) { return (n + tb - 1) / tb; };

    // init accumulators (deg starts at 1.0 for the +1 self loop)
    fill_kernel<<<blocks(N_PAD * NOUT, TB), TB, 0, stream>>>(agg, 0.f, N_PAD * NOUT);
    fill_kernel<<<blocks(N_PAD, TB), TB, 0, stream>>>(deg, 1.f, N_PAD);

    // fused edge pipeline (WMMA MLP + einsum + scatter)
    edge_kernel<<<E_ / (EW * 16), EW * 32, 0, stream>>>(
        e, eidx, x, W1, b1, W2, b2, W3, b3, agg, deg);

    rsqrt_kernel<<<blocks(N_PAD, TB), TB, 0, stream>>>(deg, dinv, N_PAD);

    // x1 = relu(agg + x@root + b)  (WMMA)
    const int nodeBlocks = N_PAD / 64;
    x1_kernel<<<nodeBlocks, 128, 0, stream>>>(x, Wr, br, agg, x1);

    // ---- GCN layer 1 ----
    gemm_node<NOUT, 32><<<nodeBlocks, 128, 0, stream>>>(x1, g1w, hw);
    fill_kernel<<<blocks(N_PAD * G_, TB), TB, 0, stream>>>(nmsg, 0.f, N_PAD * G_);
    gcn_scatter<<<blocks(E_, TB), TB, 0, stream>>>(eidx, dinv, hw, nmsg);
    gcn_finalize<<<blocks(N_ * G_, TB), TB, 0, stream>>>(nmsg, hw, dinv, g1b, x2);

    // ---- GCN layer 2 ----
    gemm_node<G_, G_><<<nodeBlocks, 128, 0, stream>>>(x2, g2w, hw);
    fill_kernel<<<blocks(N_PAD * G_, TB), TB, 0, stream>>>(nmsg, 0.f, N_PAD * G_);
    gcn_scatter<<<blocks(E_, TB), TB, 0, stream>>>(eidx, dinv, hw, nmsg);
    gcn_finalize<<<blocks(N_ * G_, TB), TB, 0, stream>>>(nmsg, hw, dinv, g2b, x3);

    // ---- pooling + head ----
    fill_kernel<<<blocks(NG * G_, TB), TB, 0, stream>>>(pooled, 0.f, NG * G_);
    pool_kernel<<<blocks(N_ * G_, TB), TB, 0, stream>>>(x3, seg, pooled);
    head_kernel<<<blocks(NG, 128), 128, 0, stream>>>(pooled, d1w, d1b, d2w, d2b, out);
}